// STSGCL_19198503813808
// MI455X (gfx1250) — compile-verified
//
#include <hip/hip_runtime.h>
#include <hip/hip_bf16.h>
#include <math.h>

typedef __attribute__((ext_vector_type(2))) float v2f;
typedef __attribute__((ext_vector_type(8))) float v8f;

// Problem constants
static constexpr int B_  = 4;
static constexpr int T_  = 12;
static constexpr int N_  = 1000;
static constexpr int C_  = 64;
static constexpr int L_  = 3;
static constexpr int TS_ = 9;
static constexpr int M4N = 4000;   // 4*N
static constexpr int NCOL = 256;   // B*C

// ---------------------------------------------------------------------------
// 1) xp = x + temporal_emb + spatial_emb   (B,T,N,C)
// ---------------------------------------------------------------------------
__global__ void add_emb(const float* __restrict__ x,
                        const float* __restrict__ te,   // (T,C)
                        const float* __restrict__ se,   // (N,C)
                        float* __restrict__ xp) {
  int id = blockIdx.x * blockDim.x + threadIdx.x;
  if (id >= B_ * T_ * N_ * C_) return;
  const int c = id & 63;
  int r = id >> 6;
  const int n = r % N_; r /= N_;
  const int t = r % T_;
  xp[id] = x[id] + te[t * C_ + c] + se[n * C_ + c];
}

// ---------------------------------------------------------------------------
// 2) gated temporal conv (taps t, t+3) -> data_res written to d_out (B,TS,N,C)
// ---------------------------------------------------------------------------
__global__ void tconv_gated(const float* __restrict__ xp,
                            const float* __restrict__ wl, const float* __restrict__ bl,
                            const float* __restrict__ wr, const float* __restrict__ br,
                            float* __restrict__ out) {
  int id = blockIdx.x * blockDim.x + threadIdx.x;
  if (id >= B_ * TS_ * N_ * C_) return;
  const int c = id & 63;
  int r = id >> 6;
  const int n = r % N_; r /= N_;
  const int ts = r % TS_;
  const int b  = r / TS_;
  const float* x0 = xp + (((size_t)b * T_ + ts) * N_ + n) * C_;       // tap at t
  const float* x3 = x0 + (size_t)3 * N_ * C_;                          // tap at t+3
  const float* wlc = wl + (size_t)c * C_ * 2;   // (C_out, C_in, 1, 2)
  const float* wrc = wr + (size_t)c * C_ * 2;
  float zl = bl[c], zr = br[c];
  #pragma unroll 8
  for (int j = 0; j < C_; ++j) {
    const float a0 = x0[j], a1 = x3[j];
    zl = fmaf(wlc[2 * j], a0, fmaf(wlc[2 * j + 1], a1, zl));
    zr = fmaf(wrc[2 * j], a0, fmaf(wrc[2 * j + 1], a1, zr));
  }
  out[id] = (1.f / (1.f + __expf(-zl))) * tanhf(zr);
}

// ---------------------------------------------------------------------------
// 3) build sliding windows: h[ts, k*N+n, b, c] = xp[b, ts+k, n, c]
//    h layout (TS, 4N, B, C) row-major
// ---------------------------------------------------------------------------
__global__ void build_wins(const float* __restrict__ xp, float* __restrict__ h) {
  int id = blockIdx.x * blockDim.x + threadIdx.x;
  if (id >= TS_ * M4N * B_ * C_) return;
  const int c = id & 63;
  const int b = (id >> 6) & 3;
  int r = id >> 8;
  const int i  = r % M4N;
  const int ts = r / M4N;
  const int k  = i / N_;
  const int n  = i % N_;
  h[id] = xp[(((size_t)b * T_ + ts + k) * N_ + n) * C_ + c];
}

// ---------------------------------------------------------------------------
// 4a) G[ts] = adj(4000x4000) @ h[ts](4000x256) via V_WMMA_F32_16X16X4_F32.
//     One wave32 computes a 32x32 C block (2x2 register tile of 16x16 WMMA
//     tiles): each A fragment is reused for 2 WMMAs, each B fragment for 2,
//     halving vector-load issue per matrix op vs a 16x16-per-wave scheme.
//     blockDim=32 -> EXEC all ones (WMMA requirement).
//     Fragment layouts per ISA 7.12.2:
//       A 16x4 : lanes 0-15 K={0,1}, lanes 16-31 K={2,3} (one K per VGPR)
//       B 4x16 : lane half selects K pair {2h,2h+1}, N = lane&15
//       C 16x16: VGPR r -> row r (lanes 0-15) / row r+8 (lanes 16-31)
// ---------------------------------------------------------------------------
__global__ __launch_bounds__(32)
void gemm_adj_h(const float* __restrict__ adj,
                const float* __restrict__ h,
                float* __restrict__ G) {
  const int lane = threadIdx.x;
  const int half = lane >> 4;      // 0 or 1
  const int lr   = lane & 15;
  const int m0 = blockIdx.x * 32;
  const int n0 = blockIdx.y * 32;
  const int ts = blockIdx.z;

  const float* Arow0 = adj + (size_t)(m0 + lr) * M4N;          // M tile 0 row
  const float* Arow1 = Arow0 + (size_t)16 * M4N;               // M tile 1 row
  const float* Hc0   = h + (size_t)ts * M4N * NCOL + n0 + lr;  // N tile 0 col
  const float* Hc1   = Hc0 + 16;                               // N tile 1 col

  v8f acc00 = {}, acc01 = {}, acc10 = {}, acc11 = {};
  #pragma unroll 2
  for (int k = 0; k < M4N; k += 4) {
    const int ka = k + 2 * half;
    // A pairs (K=ka, ka+1): contiguous, 8B aligned -> global_load_b64
    v2f a0 = *(const v2f*)(Arow0 + ka);
    v2f a1 = *(const v2f*)(Arow1 + ka);
    v2f b0, b1;
    b0[0] = Hc0[(size_t)ka * NCOL];
    b0[1] = Hc0[(size_t)(ka + 1) * NCOL];
    b1[0] = Hc1[(size_t)ka * NCOL];
    b1[1] = Hc1[(size_t)(ka + 1) * NCOL];
    acc00 = __builtin_amdgcn_wmma_f32_16x16x4_f32(false, a0, false, b0, (short)0, acc00, false, false);
    acc01 = __builtin_amdgcn_wmma_f32_16x16x4_f32(false, a0, false, b1, (short)0, acc01, false, false);
    acc10 = __builtin_amdgcn_wmma_f32_16x16x4_f32(false, a1, false, b0, (short)0, acc10, false, false);
    acc11 = __builtin_amdgcn_wmma_f32_16x16x4_f32(false, a1, false, b1, (short)0, acc11, false, false);
  }

  float* Gp = G + (size_t)ts * M4N * NCOL;
  #pragma unroll
  for (int r = 0; r < 8; ++r) {
    const size_t row0 = (size_t)(m0 + r + 8 * half) * NCOL;
    const size_t row1 = (size_t)(m0 + 16 + r + 8 * half) * NCOL;
    Gp[row0 + n0 + lr]      = acc00[r];
    Gp[row0 + n0 + 16 + lr] = acc01[r];
    Gp[row1 + n0 + lr]      = acc10[r];
    Gp[row1 + n0 + 16 + lr] = acc11[r];
  }
}

// ---------------------------------------------------------------------------
// 4b) z = G_row(64) @ W[ts,l](64x128) + b ; GLU ; write back into h;
//     running max of rows i in [N, 2N)
// ---------------------------------------------------------------------------
__global__ __launch_bounds__(64)
void glu_layer(const float* __restrict__ G,
               const float* __restrict__ gcn_w,  // (TS, L, C, 2C)
               const float* __restrict__ gcn_b,  // (TS, L, 2C)
               float* __restrict__ h,
               float* __restrict__ mx,           // (TS, N, B, C)
               int l) {
  __shared__ float g[C_];
  const int c   = threadIdx.x;     // 0..63
  const int row = blockIdx.x;      // 0..15999  ( = i*B + b )
  const int ts  = blockIdx.y;      // 0..8
  const float* Grow = G + ((size_t)ts * (M4N * B_) + row) * C_;
  g[c] = Grow[c];
  __syncthreads();

  const float* W    = gcn_w + ((size_t)ts * L_ + l) * C_ * (2 * C_);
  const float* bias = gcn_b + ((size_t)ts * L_ + l) * (2 * C_);
  float zl = bias[c], zr = bias[c + C_];
  #pragma unroll 8
  for (int j = 0; j < C_; ++j) {
    const float gv = g[j];
    zl = fmaf(gv, W[j * 128 + c], zl);
    zr = fmaf(gv, W[j * 128 + 64 + c], zr);
  }
  const float o = zl * (1.f / (1.f + __expf(-zr)));
  h[((size_t)ts * (M4N * B_) + row) * C_ + c] = o;

  const int i = row >> 2, b = row & 3;
  if (i >= N_ && i < 2 * N_) {
    float* m = mx + (((size_t)ts * N_ + (i - N_)) * B_ + b) * C_ + c;
    *m = (l == 0) ? o : fmaxf(*m, o);
  }
}

// ---------------------------------------------------------------------------
// 5) out[b,ts,n,c] += mx[ts,n,b,c]
// ---------------------------------------------------------------------------
__global__ void final_add(const float* __restrict__ mx, float* __restrict__ out) {
  int id = blockIdx.x * blockDim.x + threadIdx.x;
  if (id >= B_ * TS_ * N_ * C_) return;
  const int c = id & 63;
  int r = id >> 6;
  const int n = r % N_; r /= N_;
  const int ts = r % TS_;
  const int b  = r / TS_;
  out[id] += mx[(((size_t)ts * N_ + n) * B_ + b) * C_ + c];
}

// ---------------------------------------------------------------------------
extern "C" void kernel_launch(void* const* d_in, const int* in_sizes, int n_in,
                              void* d_out, int out_size, void* d_ws, size_t ws_size,
                              hipStream_t stream) {
  const float* x   = (const float*)d_in[0];
  const float* adj = (const float*)d_in[1];
  const float* te  = (const float*)d_in[2];
  const float* se  = (const float*)d_in[3];
  const float* wl  = (const float*)d_in[4];
  const float* bl  = (const float*)d_in[5];
  const float* wr  = (const float*)d_in[6];
  const float* br  = (const float*)d_in[7];
  const float* gw  = (const float*)d_in[8];
  const float* gb  = (const float*)d_in[9];
  float* out = (float*)d_out;

  // Workspace layout (floats)
  float* ws = (float*)d_ws;
  float* xp = ws;                                   // 3,072,000  (B,T,N,C)
  float* h  = xp + (size_t)B_ * T_ * N_ * C_;       // 9,216,000  (TS,4N,B,C)
  float* G  = h + (size_t)TS_ * M4N * NCOL;         // 9,216,000  (TS,4N,B,C)
  float* mx = G + (size_t)TS_ * M4N * NCOL;         // 2,304,000  (TS,N,B,C)

  const int nXP  = B_ * T_ * N_ * C_;
  const int nOUT = B_ * TS_ * N_ * C_;
  const int nH   = TS_ * M4N * B_ * C_;

  add_emb<<<(nXP + 255) / 256, 256, 0, stream>>>(x, te, se, xp);
  tconv_gated<<<(nOUT + 255) / 256, 256, 0, stream>>>(xp, wl, bl, wr, br, out);
  build_wins<<<(nH + 255) / 256, 256, 0, stream>>>(xp, h);

  for (int l = 0; l < L_; ++l) {
    dim3 gg(M4N / 32, NCOL / 32, TS_);   // (125, 8, 9)
    gemm_adj_h<<<gg, 32, 0, stream>>>(adj, h, G);
    dim3 g2(M4N * B_, TS_);              // (16000, 9)
    glu_layer<<<g2, 64, 0, stream>>>(G, gw, gb, h, mx, l);
  }

  final_add<<<(nOUT + 255) / 256, 256, 0, stream>>>(mx, out);
}